// GQAAttention_18021682774816
// MI455X (gfx1250) — compile-verified
//
#include <hip/hip_runtime.h>
#include <hip/hip_bf16.h>

typedef __attribute__((ext_vector_type(2))) float v2f;
typedef __attribute__((ext_vector_type(8))) float v8f;

#define N_HEADS    32
#define N_KV_HEADS 8
#define HEAD_DIM   128
#define D_MODEL    4096
#define MAX_S      8192
#define BATCH      8
#define SEQ        4
#define M_ROWS     (BATCH * SEQ)           // 32
#define KV_STRIDE  (N_KV_HEADS * HEAD_DIM) // 1024 floats between consecutive cache keys
#define ATTN_SCALE 0.08838834764831845f    // 1/sqrt(128)

__device__ __forceinline__ v8f wmma_f32(v2f a, v2f b, v8f c) {
  // D = A(16x4,f32) * B(4x16,f32) + C(16x16,f32)
  return __builtin_amdgcn_wmma_f32_16x16x4_f32(false, a, false, b, (short)0, c,
                                               false, false);
}

// ---------------------------------------------------------------------------
// Kernel 1: fused QKV projection.  x:(32,4096) -> q:(32,4096) k:(32,1024) v:(32,1024)
// One wave = one 16-wide column strip; BOTH 16-row M tiles are accumulated in
// the same wave so every weight fragment is loaded from HBM exactly once.
// ---------------------------------------------------------------------------
__global__ void qkv_gemm(const float* __restrict__ x,
                         const float* __restrict__ wq,
                         const float* __restrict__ wk,
                         const float* __restrict__ wv,
                         float* __restrict__ qb,
                         float* __restrict__ kb,
                         float* __restrict__ vb) {
  const int lane = threadIdx.x & 31;
  const int wave = threadIdx.x >> 5;
  const int nTile = blockIdx.x * 4 + wave;  // 0..383 over 6144 combined cols
  const int half = lane >> 4;
  const int l16 = lane & 15;

  const float* W;
  float* out;
  int Nw, col;
  const int nc = nTile * 16;
  if (nc < 4096)      { W = wq; Nw = 4096; col = nc;        out = qb; }
  else if (nc < 5120) { W = wk; Nw = 1024; col = nc - 4096; out = kb; }
  else                { W = wv; Nw = 1024; col = nc - 5120; out = vb; }

  const float* a0 = x + (size_t)l16 * D_MODEL + 2 * half;         // rows 0..15
  const float* a1 = x + (size_t)(16 + l16) * D_MODEL + 2 * half;  // rows 16..31
  const float* wp = W + (size_t)(2 * half) * Nw + col + l16;

  v8f c0 = {}, c1 = {};
#pragma unroll 4
  for (int k = 0; k < D_MODEL; k += 4) {
    v2f b;
    b.x = wp[0];
    b.y = wp[Nw];
    v2f va0 = *(const v2f*)a0;
    v2f va1 = *(const v2f*)a1;
    c0 = wmma_f32(va0, b, c0);
    c1 = wmma_f32(va1, b, c1);
    a0 += 4;
    a1 += 4;
    wp += (size_t)4 * Nw;
  }
  const int ccol = col + l16;
#pragma unroll
  for (int r = 0; r < 8; ++r) {
    out[(r + 8 * half) * Nw + ccol]      = c0[r];
    out[(16 + r + 8 * half) * Nw + ccol] = c1[r];
  }
}

// ---------------------------------------------------------------------------
// Kernel 2: RoPE on q (in place) and k; write roped k and raw v into the cache
// copies living in d_out at rows [pos, pos+4).  64 threads / head-vector.
// blocks: [0,1024) q heads, [1024,1280) k heads, [1280,1536) v heads.
// ---------------------------------------------------------------------------
__global__ void rope_cache(const float* __restrict__ rope_cos,
                           const float* __restrict__ rope_sin,
                           float* __restrict__ qb,
                           const float* __restrict__ kb,
                           const float* __restrict__ vb,
                           float* __restrict__ outK,
                           float* __restrict__ outV,
                           const int* __restrict__ pos_p) {
  const int blk = blockIdx.x;
  const int t = threadIdx.x;  // 0..63
  const int pos = *pos_p;

  if (blk < 1024) {  // q: (b,s,h) = 8*4*32
    const int b = blk >> 7, s = (blk >> 5) & 3, h = blk & 31;
    float* qp = qb + (b * 4 + s) * D_MODEL + h * HEAD_DIM;
    const float c = rope_cos[(pos + s) * 64 + t];
    const float sn = rope_sin[(pos + s) * 64 + t];
    const float x1 = qp[t], x2 = qp[t + 64];
    qp[t]      = x1 * c - x2 * sn;
    qp[t + 64] = x1 * sn + x2 * c;
  } else if (blk < 1280) {  // k: (b,s,h) = 8*4*8
    const int i = blk - 1024;
    const int b = i >> 5, s = (i >> 3) & 3, h = i & 7;
    const float* kp = kb + (b * 4 + s) * KV_STRIDE + h * HEAD_DIM;
    const float c = rope_cos[(pos + s) * 64 + t];
    const float sn = rope_sin[(pos + s) * 64 + t];
    const float x1 = kp[t], x2 = kp[t + 64];
    float* o = outK + (((size_t)b * MAX_S + pos + s) * N_KV_HEADS + h) * HEAD_DIM;
    o[t]      = x1 * c - x2 * sn;
    o[t + 64] = x1 * sn + x2 * c;
  } else {  // v copy into cache slice
    const int i = blk - 1280;
    const int b = i >> 5, s = (i >> 3) & 3, h = i & 7;
    const float* vp = vb + (b * 4 + s) * KV_STRIDE + h * HEAD_DIM;
    float* o = outV + (((size_t)b * MAX_S + pos + s) * N_KV_HEADS + h) * HEAD_DIM;
    o[t]      = vp[t];
    o[t + 64] = vp[t + 64];
  }
}

// ---------------------------------------------------------------------------
// Kernel 3: flash attention.  One block (4 waves, 128 thr) per (b,h).
// Q A-fragments are hoisted into registers; waves stride over 16-key blocks;
// QK^T and P*V via f32 WMMA; per-wave (m,l,acc) merged through LDS.
// ---------------------------------------------------------------------------
__global__ void attn_kernel(const float* __restrict__ qb,
                            const float* __restrict__ Kc,
                            const float* __restrict__ Vc,
                            float* __restrict__ ob,
                            const int* __restrict__ pos_p) {
  __shared__ float Pb[4][256];         // per-wave P tile (16x16), row-major
  __shared__ float accS[4][16 * 128];  // per-wave output accumulators
  __shared__ float mS[4][16];
  __shared__ float lS[4][16];

  const int b = blockIdx.x >> 5;
  const int h = blockIdx.x & 31;
  const int kvh = h >> 2;  // N_REP = 4
  const int lane = threadIdx.x & 31;
  const int wave = threadIdx.x >> 5;
  const int half = lane >> 4;
  const int l16 = lane & 15;
  const int pos = *pos_p;

  const int Lkeys = pos + SEQ;         // keys considered (4100)
  const int nblk = (Lkeys + 15) >> 4;  // 16-key blocks (257)

  // ---- hoist Q A-fragments (rows >=4 clamp to row 3; results discarded) ----
  const int mrow = (l16 < 4) ? l16 : 3;
  const float* qa = qb + ((b * 4 + mrow) * D_MODEL + h * HEAD_DIM) + 2 * half;
  v2f qfrag[HEAD_DIM / 4];
#pragma unroll
  for (int k = 0; k < HEAD_DIM / 4; ++k) qfrag[k] = *(const v2f*)(qa + 4 * k);

  const float* kbase = Kc + ((size_t)b * MAX_S * N_KV_HEADS + kvh) * HEAD_DIM;
  const float* vbase = Vc + ((size_t)b * MAX_S * N_KV_HEADS + kvh) * HEAD_DIM;

  v8f acc[8];
#pragma unroll
  for (int t = 0; t < 8; ++t) acc[t] = (v8f){};
  float mrun[8], lrun[8];
#pragma unroll
  for (int r = 0; r < 8; ++r) { mrun[r] = -3.0e38f; lrun[r] = 0.f; }

  for (int kb_i = wave; kb_i < nblk; kb_i += 4) {
    const int key0 = kb_i * 16;

    // ---- S = Q @ K^T  (16x16, f32) ----
    v8f s = {};
    const float* krow = kbase + (size_t)(key0 + l16) * KV_STRIDE + 2 * half;
#pragma unroll
    for (int k = 0; k < HEAD_DIM / 4; ++k) {
      v2f bk = *(const v2f*)(krow + 4 * k);
      s = wmma_f32(qfrag[k], bk, s);
    }

    // ---- mask + online softmax stats (rows independent) ----
    float fscale[8];
#pragma unroll
    for (int r = 0; r < 8; ++r) {
      const int row = r + 8 * half;  // query row in C layout
      const int key = key0 + l16;    // key col in C layout
      const bool ok = (key <= pos + row);
      float sv = ok ? (s[r] * ATTN_SCALE) : -3.0e38f;
      float mx = sv;
      mx = fmaxf(mx, __shfl_xor(mx, 1, 32));
      mx = fmaxf(mx, __shfl_xor(mx, 2, 32));
      mx = fmaxf(mx, __shfl_xor(mx, 4, 32));
      mx = fmaxf(mx, __shfl_xor(mx, 8, 32));
      const float mnew = fmaxf(mrun[r], mx);
      const float p = ok ? __expf(sv - mnew) : 0.f;
      float ps = p;
      ps += __shfl_xor(ps, 1, 32);
      ps += __shfl_xor(ps, 2, 32);
      ps += __shfl_xor(ps, 4, 32);
      ps += __shfl_xor(ps, 8, 32);
      const float f = __expf(mrun[r] - mnew);
      lrun[r] = lrun[r] * f + ps;
      mrun[r] = mnew;
      fscale[r] = f;
      Pb[wave][row * 16 + l16] = p;  // C layout -> row-major in LDS
    }

    // rescale running accumulator (C layout rows match fscale rows)
#pragma unroll
    for (int t = 0; t < 8; ++t)
#pragma unroll
      for (int r = 0; r < 8; ++r) acc[t][r] *= fscale[r];

    // ---- acc += P(16x16) @ V(16x128) ----
#pragma unroll
    for (int t = 0; t < 8; ++t) {
      const float* vcol = vbase + (size_t)key0 * KV_STRIDE + t * 16 + l16;
      v8f cacc = acc[t];
#pragma unroll
      for (int kk = 0; kk < 16; kk += 4) {
        v2f a = *(const v2f*)&Pb[wave][l16 * 16 + kk + 2 * half];
        v2f bv;
        bv.x = vcol[(size_t)(kk + 2 * half) * KV_STRIDE];
        bv.y = vcol[(size_t)(kk + 2 * half + 1) * KV_STRIDE];
        cacc = wmma_f32(a, bv, cacc);
      }
      acc[t] = cacc;
    }
  }

  // ---- publish per-wave state, merge across waves ----
#pragma unroll
  for (int t = 0; t < 8; ++t)
#pragma unroll
    for (int r = 0; r < 8; ++r)
      accS[wave][(r + 8 * half) * 128 + t * 16 + l16] = acc[t][r];
  if (l16 == 0) {
#pragma unroll
    for (int r = 0; r < 8; ++r) {
      mS[wave][r + 8 * half] = mrun[r];
      lS[wave][r + 8 * half] = lrun[r];
    }
  }
  __syncthreads();

  const int d = threadIdx.x;  // 0..127 = head dim
  for (int srow = 0; srow < SEQ; ++srow) {
    float m = -3.0e38f;
#pragma unroll
    for (int w = 0; w < 4; ++w) m = fmaxf(m, mS[w][srow]);
    float Lsum = 0.f, val = 0.f;
#pragma unroll
    for (int w = 0; w < 4; ++w) {
      const float f = __expf(mS[w][srow] - m);
      Lsum += f * lS[w][srow];
      val += f * accS[w][srow * 128 + d];
    }
    ob[(b * 4 + srow) * D_MODEL + h * HEAD_DIM + d] = val / Lsum;
  }
}

// ---------------------------------------------------------------------------
// Kernel 4: output projection  out = attn(32,4096) @ wo(4096,4096).
// Both M tiles fused per wave: wo streamed from HBM exactly once.
// ---------------------------------------------------------------------------
__global__ void out_gemm(const float* __restrict__ ab,
                         const float* __restrict__ wo,
                         float* __restrict__ out) {
  const int lane = threadIdx.x & 31;
  const int wave = threadIdx.x >> 5;
  const int nTile = blockIdx.x * 4 + wave;  // 0..255
  const int half = lane >> 4;
  const int l16 = lane & 15;
  const int col = nTile * 16;

  const float* a0 = ab + (size_t)l16 * D_MODEL + 2 * half;
  const float* a1 = ab + (size_t)(16 + l16) * D_MODEL + 2 * half;
  const float* wp = wo + (size_t)(2 * half) * D_MODEL + col + l16;

  v8f c0 = {}, c1 = {};
#pragma unroll 4
  for (int k = 0; k < D_MODEL; k += 4) {
    v2f b;
    b.x = wp[0];
    b.y = wp[D_MODEL];
    v2f va0 = *(const v2f*)a0;
    v2f va1 = *(const v2f*)a1;
    c0 = wmma_f32(va0, b, c0);
    c1 = wmma_f32(va1, b, c1);
    a0 += 4;
    a1 += 4;
    wp += (size_t)4 * D_MODEL;
  }
  const int ccol = col + l16;
#pragma unroll
  for (int r = 0; r < 8; ++r) {
    out[(r + 8 * half) * D_MODEL + ccol]      = c0[r];
    out[(16 + r + 8 * half) * D_MODEL + ccol] = c1[r];
  }
}

// ---------------------------------------------------------------------------
extern "C" void kernel_launch(void* const* d_in, const int* in_sizes, int n_in,
                              void* d_out, int out_size, void* d_ws, size_t ws_size,
                              hipStream_t stream) {
  const float* x        = (const float*)d_in[0];
  const float* k_cache  = (const float*)d_in[1];
  const float* v_cache  = (const float*)d_in[2];
  const float* rope_cos = (const float*)d_in[3];
  const float* rope_sin = (const float*)d_in[4];
  const float* wq       = (const float*)d_in[5];
  const float* wk       = (const float*)d_in[6];
  const float* wv       = (const float*)d_in[7];
  const float* wo       = (const float*)d_in[8];
  const int*   pos_p    = (const int*)d_in[9];  // device scalar; read in-kernel

  const size_t cacheElems = (size_t)BATCH * MAX_S * N_KV_HEADS * HEAD_DIM;  // 67108864
  float* outP = (float*)d_out;  // 32*4096
  float* outK = outP + (size_t)M_ROWS * D_MODEL;
  float* outV = outK + cacheElems;

  float* qb = (float*)d_ws;                    // 32 x 4096
  float* kb = qb + (size_t)M_ROWS * D_MODEL;   // 32 x 1024
  float* vb = kb + (size_t)M_ROWS * KV_STRIDE; // 32 x 1024
  float* ab = vb + (size_t)M_ROWS * KV_STRIDE; // 32 x 4096

  // Bulk cache copies into d_out (bandwidth-dominant part; copy-engine path).
  hipMemcpyAsync(outK, k_cache, cacheElems * sizeof(float),
                 hipMemcpyDeviceToDevice, stream);
  hipMemcpyAsync(outV, v_cache, cacheElems * sizeof(float),
                 hipMemcpyDeviceToDevice, stream);

  qkv_gemm<<<dim3(96), 128, 0, stream>>>(x, wq, wk, wv, qb, kb, vb);
  rope_cache<<<1536, 64, 0, stream>>>(rope_cos, rope_sin, qb, kb, vb, outK, outV, pos_p);
  attn_kernel<<<256, 128, 0, stream>>>(qb, outK, outV, ab, pos_p);
  out_gemm<<<dim3(64), 128, 0, stream>>>(ab, wo, outP);
}